// Wav2Vec2PretrainModel_8899172238061
// MI455X (gfx1250) — compile-verified
//
#include <hip/hip_runtime.h>
#include <math.h>

// Problem constants (match reference)
constexpr int Bc = 16, Sc = 2000, Dc = 512;
constexpr int Gc = 2, Kc = 320;
constexpr int dCV = 128;                 // per-group codevector dim
constexpr int NROWS = Bc * Sc;           // 32000
constexpr int NCOLS = Gc * Kc;           // 640
constexpr int MROWS = 32;                // rows (M) per block: two 16-row WMMA tiles
constexpr int LDA = 516;                 // padded LDS stride for A slab (bank-conflict free frags)
constexpr int LDL = 648;                 // padded LDS stride for logits slab
constexpr float EPSc = 1e-7f;

constexpr int SMEM_FLOATS = MROWS * LDL;                 // 20736 floats = 82,944 B
constexpr int SMEM_BYTES  = SMEM_FLOATS * (int)sizeof(float);

typedef float v2f __attribute__((ext_vector_type(2)));
typedef float v8f __attribute__((ext_vector_type(8)));

// ---------------------------------------------------------------------------
// Kernel 1: fused WMMA GEMM (logits) + softmax marginal + gumbel argmax +
//           codevector gather. One block = 32 rows of (B*S).
//           32-row M-tile halves W traffic through L2 vs a 16-row tile and
//           amortizes each B fragment over two WMMAs.
// ---------------------------------------------------------------------------
__global__ __launch_bounds__(256) void quantize_kernel(
    const float* __restrict__ hidden,     // (B*S, 512)
    const int*   __restrict__ mask,       // (B*S)
    const float* __restrict__ Wp,         // (640, 512)
    const float* __restrict__ bias,       // (640)
    const float* __restrict__ codevec,    // (640, 128)
    const float* __restrict__ gumbel,     // (B*S*G, 320)
    float* __restrict__ out,              // (B*S, 256) then perplexity
    float* __restrict__ gMarg)            // (640) global marginal accumulator
{
    extern __shared__ __align__(16) float smem[];  // A slab (32*516) aliased with logits (32*648)
    __shared__ float marg[NCOLS];
    __shared__ int   bestIdx[2 * MROWS];

    const int tid = threadIdx.x;
    const int r0  = blockIdx.x * MROWS;  // first global row of this block

    for (int i = tid; i < NCOLS; i += 256) marg[i] = 0.0f;

    // ---- stage A (32 rows x 512) into LDS with float4 loads ----
    for (int i = tid; i < MROWS * 128; i += 256) {
        int row = i >> 7;
        int kq  = i & 127;
        float4 v = ((const float4*)(hidden + (size_t)(r0 + row) * Dc))[kq];
        *(float4*)(&smem[row * LDA + kq * 4]) = v;
    }
    __syncthreads();

    // ---- WMMA GEMM: each wave owns 5 contiguous 16x16 N-tiles x 2 M-tiles ----
    const int lane = tid & 31;
    const int w    = tid >> 5;
    const int half = lane >> 4;          // 0: lanes 0-15, 1: lanes 16-31
    const int rr   = lane & 15;
    const int nbase = w * 5 * 16;        // first column of this wave's tiles

    v8f acc0[5], acc1[5];
    const v8f vzero = {0.f, 0.f, 0.f, 0.f, 0.f, 0.f, 0.f, 0.f};
#pragma unroll
    for (int t = 0; t < 5; ++t) { acc0[t] = vzero; acc1[t] = vzero; }

    for (int k0 = 0; k0 < Dc; k0 += 4) {
        // A frags: lane(<16): row rr, K={k0,k0+1}; lane(>=16): row rr, K={k0+2,k0+3}
        v2f a0 = *(const v2f*)(&smem[rr * LDA + k0 + 2 * half]);          // M-tile 0
        v2f a1 = *(const v2f*)(&smem[(16 + rr) * LDA + k0 + 2 * half]);   // M-tile 1
#pragma unroll
        for (int t = 0; t < 5; ++t) {
            const float* bp = Wp + (size_t)(nbase + t * 16 + rr) * Dc + k0 + 2 * half;
            v2f bf = *(const v2f*)bp;    // B = W^T: B[k][n] = W[n][k]
            acc0[t] = __builtin_amdgcn_wmma_f32_16x16x4_f32(
                false, a0, false, bf, (short)0, acc0[t], false, false);
            acc1[t] = __builtin_amdgcn_wmma_f32_16x16x4_f32(
                false, a1, false, bf, (short)0, acc1[t], false, false);
        }
    }
    __syncthreads();                     // everyone done reading the A slab

    // ---- spill logits tiles (+bias) into LDS ----
#pragma unroll
    for (int t = 0; t < 5; ++t) {
        int col = nbase + t * 16 + rr;
        float bb = bias[col];
#pragma unroll
        for (int i = 0; i < 8; ++i) {
            // C/D layout: VGPR i -> M = i + 8*half, N = lane%16
            smem[(i + 8 * half) * LDL + col]        = acc0[t][i] + bb;
            smem[(16 + i + 8 * half) * LDL + col]   = acc1[t][i] + bb;
        }
    }
    __syncthreads();

    // ---- per-(row, group) softmax stats + gumbel argmax (4 lanes per task) ----
    const int task = tid >> 2;           // 0..63
    const int sub  = tid & 3;
    const int row  = task & 31;
    const int g    = task >> 5;
    const float* L = &smem[row * LDL + g * Kc];

    float mx = -1e30f;
    for (int i = sub; i < Kc; i += 4) mx = fmaxf(mx, L[i]);
    mx = fmaxf(mx, __shfl_xor(mx, 1));
    mx = fmaxf(mx, __shfl_xor(mx, 2));

    float sum = 0.0f;
    for (int i = sub; i < Kc; i += 4) sum += expf(L[i] - mx);
    sum += __shfl_xor(sum, 1);
    sum += __shfl_xor(sum, 2);

    if (mask[r0 + row] != 0) {
        float inv = 1.0f / sum;
        for (int i = sub; i < Kc; i += 4)
            atomicAdd(&marg[g * Kc + i], expf(L[i] - mx) * inv);
    }

    // argmax of logits + gumbel (temperature is monotone -> irrelevant)
    const float* gu = gumbel + ((size_t)(r0 + row) * Gc + g) * Kc;
    float bv = -1e30f; int bi = 0;
    for (int i = sub; i < Kc; i += 4) {
        float z = L[i] + gu[i];
        if (z > bv) { bv = z; bi = i; }
    }
#pragma unroll
    for (int off = 1; off <= 2; off <<= 1) {
        float ov = __shfl_xor(bv, off);
        int   oi = __shfl_xor(bi, off);
        if (ov > bv || (ov == bv && oi < bi)) { bv = ov; bi = oi; }
    }
    if (sub == 0) bestIdx[g * MROWS + row] = bi;
    __syncthreads();

    // ---- flush per-block marginal to global accumulator ----
    for (int c = tid; c < NCOLS; c += 256)
        atomicAdd(&gMarg[c], marg[c]);

    // ---- output: gather selected codevectors, coalesced ----
    for (int idx = tid; idx < MROWS * 256; idx += 256) {
        int row2 = idx >> 8;
        int rem  = idx & 255;            // = g2*128 + j
        int g2   = rem >> 7;
        int j    = rem & 127;
        int k    = bestIdx[g2 * MROWS + row2];
        out[(size_t)(r0 + row2) * 256 + rem] =
            codevec[(size_t)(g2 * Kc + k) * dCV + j];
    }
}

// ---------------------------------------------------------------------------
// Kernel 2: perplexity from global marginal accumulator + mask count.
// ---------------------------------------------------------------------------
__global__ __launch_bounds__(256) void perplexity_kernel(
    const int* __restrict__ mask,
    const float* __restrict__ gMarg,
    float* __restrict__ out)
{
    __shared__ float rbuf[256];
    const int tid = threadIdx.x;

    // mask count
    float c = 0.0f;
    for (int i = tid; i < NROWS; i += 256) c += (mask[i] != 0) ? 1.0f : 0.0f;
    rbuf[tid] = c; __syncthreads();
    for (int s = 128; s > 0; s >>= 1) {
        if (tid < s) rbuf[tid] += rbuf[tid + s];
        __syncthreads();
    }
    float count = rbuf[0];
    __syncthreads();

    // entropy per group
    float h0 = 0.0f, h1 = 0.0f;
    for (int cc = tid; cc < NCOLS; cc += 256) {
        float p = gMarg[cc] / count;
        float term = p * logf(p + EPSc);
        if (cc < Kc) h0 += term; else h1 += term;
    }
    rbuf[tid] = h0; __syncthreads();
    for (int s = 128; s > 0; s >>= 1) {
        if (tid < s) rbuf[tid] += rbuf[tid + s];
        __syncthreads();
    }
    float H0 = rbuf[0];
    __syncthreads();
    rbuf[tid] = h1; __syncthreads();
    for (int s = 128; s > 0; s >>= 1) {
        if (tid < s) rbuf[tid] += rbuf[tid + s];
        __syncthreads();
    }
    float H1 = rbuf[0];

    if (tid == 0)
        out[(size_t)NROWS * 256] = expf(-H0) + expf(-H1);
}

// ---------------------------------------------------------------------------
extern "C" void kernel_launch(void* const* d_in, const int* in_sizes, int n_in,
                              void* d_out, int out_size, void* d_ws, size_t ws_size,
                              hipStream_t stream) {
    const float* hidden  = (const float*)d_in[0];
    const int*   mask    = (const int*)d_in[1];
    const float* Wp      = (const float*)d_in[2];
    const float* bias    = (const float*)d_in[3];
    const float* codevec = (const float*)d_in[4];
    const float* gumbel  = (const float*)d_in[5];
    float* out   = (float*)d_out;
    float* gMarg = (float*)d_ws;

    // Allow >64KB dynamic LDS (CDNA5 WGP has 320KB). Host-side, non-stream,
    // deterministic -> graph-capture safe.
    (void)hipFuncSetAttribute((const void*)quantize_kernel,
                              hipFuncAttributeMaxDynamicSharedMemorySize,
                              SMEM_BYTES);

    hipMemsetAsync(gMarg, 0, NCOLS * sizeof(float), stream);

    quantize_kernel<<<NROWS / MROWS, 256, SMEM_BYTES, stream>>>(
        hidden, mask, Wp, bias, codevec, gumbel, out, gMarg);

    perplexity_kernel<<<1, 256, 0, stream>>>(mask, gMarg, out);
}